// LinkConvInPillar_44092134261325
// MI455X (gfx1250) — compile-verified
//
#include <hip/hip_runtime.h>
#include <hip/hip_bf16.h>

typedef float v2f __attribute__((ext_vector_type(2)));
typedef float v8f __attribute__((ext_vector_type(8)));

#define BN_EPS 1e-3f

// -------- workspace layout (float offsets) --------
#define WS_SUM1 0
#define WS_SQ1  128
#define WS_SUM2 256
#define WS_SQ2  384
#define WS_A1   512
#define WS_B1   640
#define WS_A2   768
#define WS_B2   896
#define WS_SEG  1024   // seg[G][128] lives here

// ---------------------------------------------------------------------------
// Zero the accumulation regions (stats + segment buffer). G read on-device so
// this stays graph-capture safe.
// ---------------------------------------------------------------------------
__global__ void k_zero_ws(float* __restrict__ ws, const int* __restrict__ d_G) {
    long long n = 1024LL + 128LL * (long long)(*d_G);
    long long stride = (long long)gridDim.x * blockDim.x;
    for (long long i = (long long)blockIdx.x * blockDim.x + threadIdx.x; i < n; i += stride)
        ws[i] = 0.0f;
}

// ---------------------------------------------------------------------------
// Core 16(M) x 16(N) x 64(K) f32 tile via 16x V_WMMA_F32_16X16X4_F32.
// A frags preloaded (a[kk].x = A[row][4kk+2hi], a[kk].y = A[row][4kk+2hi+1],
// row = rowBase + (lane&15)).  wcol = W + col + 256*hi, col = 16*nt + (lane&15).
// Two accumulator chains break the WMMA->WMMA RAW dependency.
// ---------------------------------------------------------------------------
__device__ __forceinline__ v8f gemm_tile(const v2f* __restrict__ a,
                                         const float* __restrict__ wcol) {
    v8f acc0 = {0.f,0.f,0.f,0.f,0.f,0.f,0.f,0.f};
    v8f acc1 = {0.f,0.f,0.f,0.f,0.f,0.f,0.f,0.f};
#pragma unroll
    for (int kk = 0; kk < 16; kk += 2) {
        v2f b0, b1;
        b0.x = wcol[512 * kk];
        b0.y = wcol[512 * kk + 128];
        b1.x = wcol[512 * (kk + 1)];
        b1.y = wcol[512 * (kk + 1) + 128];
        acc0 = __builtin_amdgcn_wmma_f32_16x16x4_f32(false, a[kk],     false, b0,
                                                     (short)0, acc0, false, false);
        acc1 = __builtin_amdgcn_wmma_f32_16x16x4_f32(false, a[kk + 1], false, b1,
                                                     (short)0, acc1, false, false);
    }
    return acc0 + acc1;
}

__device__ __forceinline__ void load_afrags(const float* __restrict__ feat,
                                            int row, int hi, v2f* a) {
    const float* arow = feat + (long long)row * 64 + 2 * hi;
#pragma unroll
    for (int kk = 0; kk < 16; ++kk)
        a[kk] = *(const v2f*)(arow + 4 * kk);
}

// ---------------------------------------------------------------------------
// Pass 1: GEMM, accumulate per-channel sum / sumsq of z (bias omitted — it
// cancels inside BN).
// ---------------------------------------------------------------------------
__global__ void __launch_bounds__(256)
k_gemm_stats1(const float* __restrict__ feat, const float* __restrict__ W,
              float* __restrict__ ws, int n) {
    __shared__ float s_sum[128];
    __shared__ float s_sq[128];
    int tid = threadIdx.x;
    if (tid < 128) s_sum[tid] = 0.0f; else s_sq[tid - 128] = 0.0f;
    __syncthreads();

    int wave = tid >> 5, lane = tid & 31;
    int l = lane & 15, hi = lane >> 4;
    int rowBase = blockIdx.x * 128 + wave * 16;
    int rowA = rowBase + l; if (rowA >= n) rowA = n - 1;

    v2f a[16];
    load_afrags(feat, rowA, hi, a);

    for (int nt = 0; nt < 8; ++nt) {
        int col = 16 * nt + l;
        v8f acc = gemm_tile(a, W + col + 256 * hi);
        float s = 0.0f, q = 0.0f;
#pragma unroll
        for (int r = 0; r < 8; ++r) {
            int rr = rowBase + r + 8 * hi;
            float v = (rr < n) ? acc[r] : 0.0f;
            s += v; q += v * v;
        }
        s += __shfl_xor(s, 16, 32);
        q += __shfl_xor(q, 16, 32);
        if (hi == 0) { atomicAdd(&s_sum[col], s); atomicAdd(&s_sq[col], q); }
    }
    __syncthreads();
    if (tid < 128) {
        atomicAdd(&ws[WS_SUM1 + tid], s_sum[tid]);
        atomicAdd(&ws[WS_SQ1 + tid],  s_sq[tid]);
    }
}

// ---------------------------------------------------------------------------
// Fold BN stats into per-channel affine: alpha = g*rsqrt(var+eps),
// beta = be - mean*alpha.  One 128-thread block.
// ---------------------------------------------------------------------------
__global__ void k_finalize(const float* __restrict__ g, const float* __restrict__ be,
                           float* __restrict__ ws, int sumOff, int sqOff,
                           int aOff, int bOff, float invN) {
    int c = threadIdx.x;
    float mean = ws[sumOff + c] * invN;
    float var  = ws[sqOff + c] * invN - mean * mean;
    float al   = g[c] * rsqrtf(var + BN_EPS);
    ws[aOff + c] = al;
    ws[bOff + c] = be[c] - mean * al;
}

// ---------------------------------------------------------------------------
// Pass 2: GEMM -> feat, pw1 from floor(xyz); scatter pw1*feat into seg[g][c].
// Row index is uniform per half-wave -> 16 contiguous atomics per half-wave.
// ---------------------------------------------------------------------------
__global__ void __launch_bounds__(256)
k_scatter(const float* __restrict__ feat, const float* __restrict__ W,
          const float* __restrict__ xyz, const int* __restrict__ inv,
          const float* __restrict__ Wp1, const float* __restrict__ bp1,
          float* __restrict__ ws, int n) {
    int tid = threadIdx.x;
    int wave = tid >> 5, lane = tid & 31;
    int l = lane & 15, hi = lane >> 4;
    int rowBase = blockIdx.x * 128 + wave * 16;
    int rowA = rowBase + l; if (rowA >= n) rowA = n - 1;

    v2f a[16];
    load_afrags(feat, rowA, hi, a);

    float px[8], py[8], pz[8];
    int   gidx[8];
    bool  valid[8];
#pragma unroll
    for (int r = 0; r < 8; ++r) {
        int rr = rowBase + r + 8 * hi;
        valid[r] = rr < n;
        int rc = valid[r] ? rr : 0;
        px[r] = floorf(xyz[rc * 3 + 0]);
        py[r] = floorf(xyz[rc * 3 + 1]);
        pz[r] = floorf(xyz[rc * 3 + 2]);
        gidx[r] = inv[rc];
    }

    float* seg = ws + WS_SEG;
    for (int nt = 0; nt < 8; ++nt) {
        int col = 16 * nt + l;
        v8f acc = gemm_tile(a, W + col + 256 * hi);
        float al = ws[WS_A1 + col], bt = ws[WS_B1 + col];
        float w0 = Wp1[col], w1 = Wp1[128 + col], w2 = Wp1[256 + col], bb = bp1[col];
#pragma unroll
        for (int r = 0; r < 8; ++r) {
            if (valid[r]) {
                float f  = acc[r] * al + bt;
                float pw = px[r] * w0 + py[r] * w1 + pz[r] * w2 + bb;
                atomicAdd(&seg[(long long)gidx[r] * 128 + col], pw * f);
            }
        }
    }
}

// ---------------------------------------------------------------------------
// Pass 3: GEMM -> feat, pw2; o = pw2*feat - seg[g][c]; write pre-BN output,
// accumulate BN2 stats.
// ---------------------------------------------------------------------------
__global__ void __launch_bounds__(256)
k_gather_out(const float* __restrict__ feat, const float* __restrict__ W,
             const float* __restrict__ xyz, const int* __restrict__ inv,
             const float* __restrict__ Wp2, const float* __restrict__ bp2,
             float* __restrict__ ws, float* __restrict__ out, int n) {
    __shared__ float s_sum[128];
    __shared__ float s_sq[128];
    int tid = threadIdx.x;
    if (tid < 128) s_sum[tid] = 0.0f; else s_sq[tid - 128] = 0.0f;
    __syncthreads();

    int wave = tid >> 5, lane = tid & 31;
    int l = lane & 15, hi = lane >> 4;
    int rowBase = blockIdx.x * 128 + wave * 16;
    int rowA = rowBase + l; if (rowA >= n) rowA = n - 1;

    v2f a[16];
    load_afrags(feat, rowA, hi, a);

    float px[8], py[8], pz[8];
    int   gidx[8];
    bool  valid[8];
#pragma unroll
    for (int r = 0; r < 8; ++r) {
        int rr = rowBase + r + 8 * hi;
        valid[r] = rr < n;
        int rc = valid[r] ? rr : 0;
        px[r] = floorf(xyz[rc * 3 + 0]);
        py[r] = floorf(xyz[rc * 3 + 1]);
        pz[r] = floorf(xyz[rc * 3 + 2]);
        gidx[r] = inv[rc];
    }

    const float* seg = ws + WS_SEG;
    for (int nt = 0; nt < 8; ++nt) {
        int col = 16 * nt + l;
        v8f acc = gemm_tile(a, W + col + 256 * hi);
        float al = ws[WS_A1 + col], bt = ws[WS_B1 + col];
        float w0 = Wp2[col], w1 = Wp2[128 + col], w2 = Wp2[256 + col], bb = bp2[col];
        float s = 0.0f, q = 0.0f;
#pragma unroll
        for (int r = 0; r < 8; ++r) {
            if (valid[r]) {
                int   rr = rowBase + r + 8 * hi;
                float f  = acc[r] * al + bt;
                float pw = px[r] * w0 + py[r] * w1 + pz[r] * w2 + bb;
                float o  = pw * f - seg[(long long)gidx[r] * 128 + col];
                out[(long long)rr * 128 + col] = o;
                s += o; q += o * o;
            }
        }
        s += __shfl_xor(s, 16, 32);
        q += __shfl_xor(q, 16, 32);
        if (hi == 0) { atomicAdd(&s_sum[col], s); atomicAdd(&s_sq[col], q); }
    }
    __syncthreads();
    if (tid < 128) {
        atomicAdd(&ws[WS_SUM2 + tid], s_sum[tid]);
        atomicAdd(&ws[WS_SQ2 + tid],  s_sq[tid]);
    }
}

// ---------------------------------------------------------------------------
// In-place BN2 affine + ReLU, vectorized float4 (channels 4-aligned, no wrap).
// ---------------------------------------------------------------------------
__global__ void k_bn_relu(float* __restrict__ out, const float* __restrict__ ws,
                          long long total4) {
    long long i = (long long)blockIdx.x * blockDim.x + threadIdx.x;
    if (i >= total4) return;
    int c0 = (int)((i * 4) & 127);
    float4 v = ((float4*)out)[i];
    v.x = fmaxf(v.x * ws[WS_A2 + c0 + 0] + ws[WS_B2 + c0 + 0], 0.0f);
    v.y = fmaxf(v.y * ws[WS_A2 + c0 + 1] + ws[WS_B2 + c0 + 1], 0.0f);
    v.z = fmaxf(v.z * ws[WS_A2 + c0 + 2] + ws[WS_B2 + c0 + 2], 0.0f);
    v.w = fmaxf(v.w * ws[WS_A2 + c0 + 3] + ws[WS_B2 + c0 + 3], 0.0f);
    ((float4*)out)[i] = v;
}

// ---------------------------------------------------------------------------
extern "C" void kernel_launch(void* const* d_in, const int* in_sizes, int n_in,
                              void* d_out, int out_size, void* d_ws, size_t ws_size,
                              hipStream_t stream) {
    const float* xyz   = (const float*)d_in[0];
    const float* featA = (const float*)d_in[1];
    const int*   inv   = (const int*)d_in[2];
    const float* W_pre = (const float*)d_in[3];
    // d_in[4] = b_pre: cancels inside BN, unused.
    const float* g1    = (const float*)d_in[5];
    const float* be1   = (const float*)d_in[6];
    const float* Wp1   = (const float*)d_in[7];
    const float* bp1   = (const float*)d_in[8];
    const float* Wp2   = (const float*)d_in[9];
    const float* bp2   = (const float*)d_in[10];
    const float* g2    = (const float*)d_in[11];
    const float* be2   = (const float*)d_in[12];
    const int*   d_G   = (const int*)d_in[13];

    float* ws  = (float*)d_ws;
    float* out = (float*)d_out;

    int n = in_sizes[0] / 3;                 // number of points
    float invN = 1.0f / (float)n;
    int gemmBlocks = (n + 127) / 128;
    long long total4 = (long long)n * 32;    // n*128/4
    int reluBlocks = (int)((total4 + 255) / 256);

    k_zero_ws<<<2048, 256, 0, stream>>>(ws, d_G);

    k_gemm_stats1<<<gemmBlocks, 256, 0, stream>>>(featA, W_pre, ws, n);
    k_finalize<<<1, 128, 0, stream>>>(g1, be1, ws, WS_SUM1, WS_SQ1, WS_A1, WS_B1, invN);

    k_scatter<<<gemmBlocks, 256, 0, stream>>>(featA, W_pre, xyz, inv, Wp1, bp1, ws, n);

    k_gather_out<<<gemmBlocks, 256, 0, stream>>>(featA, W_pre, xyz, inv, Wp2, bp2, ws, out, n);
    k_finalize<<<1, 128, 0, stream>>>(g2, be2, ws, WS_SUM2, WS_SQ2, WS_A2, WS_B2, invN);

    k_bn_relu<<<reluBlocks, 256, 0, stream>>>(out, ws, total4);
}